// KDHR_86380382257341
// MI455X (gfx1250) — compile-verified
//
#include <hip/hip_runtime.h>
#include <hip/hip_bf16.h>

#define NSH 1195
#define NSS 390
#define NHH 805
#define DEMB 64
#define HDIM 256
#define NB 20000
#define ESH 1000000
#define ESS 200000
#define EHH 200000
#define BN_EPS 1e-5f

typedef __attribute__((ext_vector_type(2))) float v2f;
typedef __attribute__((ext_vector_type(8))) float v8f;

__device__ __forceinline__ v8f wmma4(v2f a, v2f b, v8f c) {
  // D = A(16x4,f32) * B(4x16,f32) + C(16x16,f32)
  return __builtin_amdgcn_wmma_f32_16x16x4_f32(false, a, false, b, (short)0, c,
                                               false, false);
}

__device__ __forceinline__ int imin(int a, int b) { return a < b ? a : b; }

// ---------------- elementwise / utility kernels ----------------

__global__ void zero_k(float* p, long long n) {
  long long i = (long long)blockIdx.x * blockDim.x + threadIdx.x;
  if (i < n) p[i] = 0.f;
}

__global__ void gather_rows_k(const float* __restrict__ src, const int* __restrict__ idx,
                              float* __restrict__ out, int n, int F) {
  long long i = (long long)blockIdx.x * blockDim.x + threadIdx.x;
  if (i >= (long long)n * F) return;
  int r = (int)(i / F), c = (int)(i % F);
  out[i] = src[(long long)idx[r] * F + c];
}

__global__ void concat_hh_k(const float* __restrict__ emb, const int* __restrict__ idx,
                            const float* __restrict__ kg, float* __restrict__ out, int n) {
  long long i = (long long)blockIdx.x * blockDim.x + threadIdx.x;
  if (i >= (long long)n * 91) return;
  int r = (int)(i / 91), c = (int)(i % 91);
  out[i] = (c < 64) ? emb[(long long)idx[r] * 64 + c] : kg[r * 27 + (c - 64)];
}

__global__ void deg_k(const int* __restrict__ ei, int E, float* __restrict__ cnt) {
  int e = blockIdx.x * blockDim.x + threadIdx.x;
  if (e >= E) return;
  atomicAdd(cnt + ei[E + e], 1.0f);
}

// agg[dst] += h[src], 4 floats per thread
__global__ void scatter_add4_k(const float* __restrict__ h, const int* __restrict__ ei,
                               int E, int F4, float* __restrict__ agg) {
  long long tid = (long long)blockIdx.x * blockDim.x + threadIdx.x;
  if (tid >= (long long)E * F4) return;
  int e = (int)(tid / F4);
  int g = (int)(tid % F4);
  int src = ei[e];
  int dst = ei[E + e];
  int F = F4 * 4;
  const float4 v = *(const float4*)(h + (long long)src * F + g * 4);
  float* p = agg + (long long)dst * F + g * 4;
  atomicAdd(p + 0, v.x);
  atomicAdd(p + 1, v.y);
  atomicAdd(p + 2, v.z);
  atomicAdd(p + 3, v.w);
}

__global__ void mean_tanh_k(const float* __restrict__ agg, const float* __restrict__ cnt,
                            float* __restrict__ out, int n, int F) {
  long long i = (long long)blockIdx.x * blockDim.x + threadIdx.x;
  if (i >= (long long)n * F) return;
  int r = (int)(i / F);
  out[i] = tanhf(agg[i] / fmaxf(cnt[r], 1.0f));
}

__global__ void tanh_inplace_k(float* x, long long n) {
  long long i = (long long)blockIdx.x * blockDim.x + threadIdx.x;
  if (i < n) x[i] = tanhf(x[i]);
}

__global__ void avg3_k(const float* a, const float* b, const float* c, float* o, long long n) {
  long long i = (long long)blockIdx.x * blockDim.x + threadIdx.x;
  if (i < n) o[i] = (a[i] + b[i] + c[i]) * (1.0f / 3.0f);
}

__global__ void add_k(const float* a, const float* b, float* o, long long n) {
  long long i = (long long)blockIdx.x * blockDim.x + threadIdx.x;
  if (i < n) o[i] = a[i] + b[i];
}

// y[N,Dout] = x[N,Din] @ W[Din,Dout] + b
__global__ void lin_k(const float* __restrict__ x, const float* __restrict__ W,
                      const float* __restrict__ b, float* __restrict__ y,
                      int N, int Din, int Dout) {
  long long tid = (long long)blockIdx.x * blockDim.x + threadIdx.x;
  if (tid >= (long long)N * Dout) return;
  int i = (int)(tid / Dout), j = (int)(tid % Dout);
  float acc = b[j];
  const float* xr = x + (long long)i * Din;
  for (int k = 0; k < Din; ++k) acc += xr[k] * W[(long long)k * Dout + j];
  y[tid] = acc;
}

// one wave per row
__global__ void rowsum_k(const float* __restrict__ P, float* __restrict__ out, int M, int K) {
  int row = blockIdx.x * (blockDim.x >> 5) + (threadIdx.x >> 5);
  if (row >= M) return;
  int lane = threadIdx.x & 31;
  float s = 0.f;
  const float* pr = P + (long long)row * K;
  for (int k = lane; k < K; k += 32) s += pr[k];
  for (int off = 16; off > 0; off >>= 1) s += __shfl_down(s, off, 32);
  if (lane == 0) out[row] = s;
}

// per-column batch stats (biased var), one block per column
__global__ void bn_stats_k(const float* __restrict__ x, int N, int C,
                           float* __restrict__ mean, float* __restrict__ rstd) {
  int c = blockIdx.x;
  float s = 0.f, s2 = 0.f;
  for (int r = threadIdx.x; r < N; r += blockDim.x) {
    float v = x[(long long)r * C + c];
    s += v;
    s2 += v * v;
  }
  __shared__ float sh1[256], sh2[256];
  sh1[threadIdx.x] = s;
  sh2[threadIdx.x] = s2;
  __syncthreads();
  for (int off = blockDim.x >> 1; off > 0; off >>= 1) {
    if ((int)threadIdx.x < off) {
      sh1[threadIdx.x] += sh1[threadIdx.x + off];
      sh2[threadIdx.x] += sh2[threadIdx.x + off];
    }
    __syncthreads();
  }
  if (threadIdx.x == 0) {
    float m = sh1[0] / (float)N;
    float var = sh2[0] / (float)N - m * m;
    mean[c] = m;
    rstd[c] = rsqrtf(var + BN_EPS);
  }
}

// x = act((x - mean) * rstd * g + be), act = tanh (mode 0) or relu (mode 1)
__global__ void bn_apply_k(float* __restrict__ x, const float* __restrict__ mean,
                           const float* __restrict__ rstd, const float* __restrict__ g,
                           const float* __restrict__ be, long long n, int C, int relu) {
  long long i = (long long)blockIdx.x * blockDim.x + threadIdx.x;
  if (i >= n) return;
  int c = (int)(i % C);
  float v = (x[i] - mean[c]) * rstd[c] * g[c] + be[c];
  x[i] = relu ? fmaxf(v, 0.f) : tanhf(v);
}

// ---------------- WMMA GEMM kernels (f32 16x16x4) ----------------
// One wave computes a 16x64 output strip (4 accumulators sharing one A tile).
// 8 waves / 256-thread block. All main-loop loads are unguarded (branchless).

// Out[M,N] = (P[M,K] @ Q[K,N]) / rsum[row] ; M%16==0, N%64==0, K arbitrary
__global__ void gemm_scaled_k(const float* __restrict__ P, const float* __restrict__ Q,
                              const float* __restrict__ rsum, float* __restrict__ Out,
                              int M, int N, int K) {
  int wid = (blockIdx.x * blockDim.x + threadIdx.x) >> 5;
  int groupsN = N >> 6;
  int tileM = wid / groupsN, g = wid % groupsN;
  if (tileM >= (M >> 4)) return;
  int lane = threadIdx.x & 31;
  int mn = lane & 15;
  int hi = lane >> 4;
  int rowA = tileM * 16 + mn;
  int colBase = g * 64 + mn;
  const float* pr = P + (long long)rowA * K;
  v8f acc[4] = {};
  const int K4 = K & ~3;
  for (int k = 0; k < K4; k += 4) {
    int k0 = k + 2 * hi;
    v2f a = *(const v2f*)(pr + k0);
    const float* q0 = Q + (long long)k0 * N + colBase;
#pragma unroll
    for (int t = 0; t < 4; ++t) {
      v2f b;
      b.x = q0[16 * t];
      b.y = q0[N + 16 * t];
      acc[t] = wmma4(a, b, acc[t]);
    }
  }
  if (K4 < K) {  // K tail: clamped addresses + branchless select
    int k0 = K4 + 2 * hi;
    int kc0 = imin(k0, K - 1), kc1 = imin(k0 + 1, K - 1);
    bool ok0 = k0 < K, ok1 = (k0 + 1) < K;
    v2f a;
    a.x = ok0 ? pr[kc0] : 0.f;
    a.y = ok1 ? pr[kc1] : 0.f;
    const float* q0 = Q + (long long)kc0 * N + colBase;
    const float* q1 = Q + (long long)kc1 * N + colBase;
#pragma unroll
    for (int t = 0; t < 4; ++t) {
      v2f b;
      b.x = ok0 ? q0[16 * t] : 0.f;
      b.y = ok1 ? q1[16 * t] : 0.f;
      acc[t] = wmma4(a, b, acc[t]);
    }
  }
#pragma unroll
  for (int v = 0; v < 8; ++v) {
    int row = tileM * 16 + v + 8 * hi;
    float rs = 1.0f / rsum[row];
#pragma unroll
    for (int t = 0; t < 4; ++t)
      Out[(long long)row * N + colBase + 16 * t] = acc[t][v] * rs;
  }
}

// Out[M,N] = P[M,K] @ Q[K,N] + bias[col] ; M%16==0, N%64==0, K%4==0
__global__ void gemm_bias_k(const float* __restrict__ P, const float* __restrict__ Q,
                            const float* __restrict__ bias, float* __restrict__ Out,
                            int M, int N, int K) {
  int wid = (blockIdx.x * blockDim.x + threadIdx.x) >> 5;
  int groupsN = N >> 6;
  int tileM = wid / groupsN, g = wid % groupsN;
  if (tileM >= (M >> 4)) return;
  int lane = threadIdx.x & 31;
  int mn = lane & 15;
  int hi = lane >> 4;
  int rowA = tileM * 16 + mn;
  int colBase = g * 64 + mn;
  const float* pr = P + (long long)rowA * K;
  v8f acc[4] = {};
  for (int k = 0; k < K; k += 4) {
    int k0 = k + 2 * hi;
    v2f a = *(const v2f*)(pr + k0);
    const float* q0 = Q + (long long)k0 * N + colBase;
#pragma unroll
    for (int t = 0; t < 4; ++t) {
      v2f b;
      b.x = q0[16 * t];
      b.y = q0[N + 16 * t];
      acc[t] = wmma4(a, b, acc[t]);
    }
  }
#pragma unroll
  for (int v = 0; v < 8; ++v) {
    int row = tileM * 16 + v + 8 * hi;
#pragma unroll
    for (int t = 0; t < 4; ++t)
      Out[(long long)row * N + colBase + 16 * t] = acc[t][v] + bias[colBase + 16 * t];
  }
}

// Out[M,N] = P[M,K] @ Qt[N,K]^T ; M%16==0, K%4==0, N arbitrary.
// OOB columns: addresses clamped (garbage accumulated, never stored).
__global__ void gemm_bt_k(const float* __restrict__ P, const float* __restrict__ Qt,
                          float* __restrict__ Out, int M, int N, int K) {
  int wid = (blockIdx.x * blockDim.x + threadIdx.x) >> 5;
  int groupsN = (N + 63) >> 6;
  int tileM = wid / groupsN, g = wid % groupsN;
  if (tileM >= (M >> 4)) return;
  int lane = threadIdx.x & 31;
  int mn = lane & 15;
  int hi = lane >> 4;
  int rowA = tileM * 16 + mn;
  int colBase = g * 64 + mn;
  const float* pr = P + (long long)rowA * K;
  const float* qr[4];
#pragma unroll
  for (int t = 0; t < 4; ++t)
    qr[t] = Qt + (long long)imin(colBase + 16 * t, N - 1) * K;
  v8f acc[4] = {};
  for (int k = 0; k < K; k += 4) {
    int k0 = k + 2 * hi;
    v2f a = *(const v2f*)(pr + k0);
#pragma unroll
    for (int t = 0; t < 4; ++t) {
      v2f b = *(const v2f*)(qr[t] + k0);
      acc[t] = wmma4(a, b, acc[t]);
    }
  }
#pragma unroll
  for (int t = 0; t < 4; ++t) {
    int cn = colBase + 16 * t;
    if (cn < N) {
#pragma unroll
      for (int v = 0; v < 8; ++v) {
        int row = tileM * 16 + v + 8 * hi;
        Out[(long long)row * N + cn] = acc[t][v];
      }
    }
  }
}

// ---------------- host-side orchestration ----------------

static inline unsigned NBLK(long long n, int t) { return (unsigned)((n + t - 1) / t); }

extern "C" void kernel_launch(void* const* d_in, const int* in_sizes, int n_in,
                              void* d_out, int out_size, void* d_ws, size_t ws_size,
                              hipStream_t stream) {
  const int* xSH = (const int*)d_in[0];
  const int* eiSH = (const int*)d_in[1];
  const int* xSS = (const int*)d_in[2];
  const int* eiSS = (const int*)d_in[3];
  const int* xHH = (const int*)d_in[4];
  const int* eiHH = (const int*)d_in[5];
  const float* presc = (const float*)d_in[6];
  const float* kg = (const float*)d_in[7];
  const float* emb = (const float*)d_in[8];
  const float* w_sh1 = (const float*)d_in[9];
  const float* b_sh1 = (const float*)d_in[10];
  const float* w_sh2 = (const float*)d_in[11];
  const float* b_sh2 = (const float*)d_in[12];
  const float* w_mlp1 = (const float*)d_in[13];
  const float* b_mlp1 = (const float*)d_in[14];
  const float* g_bn1 = (const float*)d_in[15];
  const float* be_bn1 = (const float*)d_in[16];
  const float* w_sh1h = (const float*)d_in[17];
  const float* b_sh1h = (const float*)d_in[18];
  const float* w_sh2h = (const float*)d_in[19];
  const float* b_sh2h = (const float*)d_in[20];
  const float* w_mlp1h = (const float*)d_in[21];
  const float* b_mlp1h = (const float*)d_in[22];
  const float* g_bn1h = (const float*)d_in[23];
  const float* be_bn1h = (const float*)d_in[24];
  const float* w_ss = (const float*)d_in[25];
  const float* b_ss = (const float*)d_in[26];
  const float* w_hh = (const float*)d_in[27];
  const float* b_hh = (const float*)d_in[28];
  const float* w_mlp = (const float*)d_in[29];
  const float* b_mlp = (const float*)d_in[30];
  const float* g_si = (const float*)d_in[31];
  const float* be_si = (const float*)d_in[32];
  float* out = (float*)d_out;

  float* ws = (float*)d_ws;
  size_t o = 0;
  auto alloc = [&](size_t n) { float* p = ws + o; o += (n + 7) & ~(size_t)7; return p; };

  float* x1   = alloc((size_t)NSH * DEMB);
  float* x2   = alloc((size_t)NSH * DEMB);
  float* x6   = alloc((size_t)NSH * DEMB);
  float* x22  = alloc((size_t)NSH * DEMB);
  float* x66  = alloc((size_t)NSH * DEMB);
  float* x9   = alloc((size_t)NSH * DEMB);
  float* x99  = alloc((size_t)NSH * DEMB);
  float* h64  = alloc((size_t)NSH * DEMB);
  float* agg64 = alloc((size_t)NSH * DEMB);
  float* cnt  = alloc((size_t)NSH);
  float* y9   = alloc((size_t)NSH * HDIM);
  float* y99  = alloc((size_t)NSH * HDIM);
  float* xss0 = alloc((size_t)NSS * DEMB);
  float* hss  = alloc((size_t)NSS * HDIM);
  float* aggss = alloc((size_t)NSS * HDIM);
  float* xhh0 = alloc((size_t)NHH * 91);
  float* hhh  = alloc((size_t)NHH * HDIM);
  float* agghh = alloc((size_t)NHH * HDIM);
  float* es   = alloc((size_t)NSS * HDIM);
  float* eh   = alloc((size_t)NHH * HDIM);
  float* psum = alloc((size_t)NB);
  float* e0   = alloc((size_t)NB * HDIM);
  float* y2   = alloc((size_t)NB * HDIM);
  float* bmean = alloc(256);
  float* brstd = alloc(256);

  const int T = 256;
  const long long nSH64 = (long long)NSH * DEMB;
  const long long nSH256 = (long long)NSH * HDIM;

  // x1 = emb[x_SH]
  gather_rows_k<<<NBLK(nSH64, T), T, 0, stream>>>(emb, xSH, x1, NSH, DEMB);

  // SH in-degree (shared by all 4 mean-GCN passes)
  zero_k<<<NBLK(NSH, T), T, 0, stream>>>(cnt, NSH);
  deg_k<<<NBLK(ESH, T), T, 0, stream>>>(eiSH, ESH, cnt);

  // ---- symptom branch: two mean-GCN layers on SH ----
  lin_k<<<NBLK(nSH64, T), T, 0, stream>>>(x1, w_sh1, b_sh1, h64, NSH, DEMB, DEMB);
  zero_k<<<NBLK(nSH64, T), T, 0, stream>>>(agg64, nSH64);
  scatter_add4_k<<<NBLK((long long)ESH * 16, T), T, 0, stream>>>(h64, eiSH, ESH, 16, agg64);
  mean_tanh_k<<<NBLK(nSH64, T), T, 0, stream>>>(agg64, cnt, x2, NSH, DEMB);

  lin_k<<<NBLK(nSH64, T), T, 0, stream>>>(x2, w_sh2, b_sh2, h64, NSH, DEMB, DEMB);
  zero_k<<<NBLK(nSH64, T), T, 0, stream>>>(agg64, nSH64);
  scatter_add4_k<<<NBLK((long long)ESH * 16, T), T, 0, stream>>>(h64, eiSH, ESH, 16, agg64);
  mean_tanh_k<<<NBLK(nSH64, T), T, 0, stream>>>(agg64, cnt, x6, NSH, DEMB);

  avg3_k<<<NBLK(nSH64, T), T, 0, stream>>>(x1, x2, x6, x9, nSH64);
  lin_k<<<NBLK(nSH256, T), T, 0, stream>>>(x9, w_mlp1, b_mlp1, y9, NSH, DEMB, HDIM);
  bn_stats_k<<<HDIM, 256, 0, stream>>>(y9, NSH, HDIM, bmean, brstd);
  bn_apply_k<<<NBLK(nSH256, T), T, 0, stream>>>(y9, bmean, brstd, g_bn1, be_bn1, nSH256, HDIM, 0);

  // ---- herb branch: two mean-GCN layers on SH ----
  lin_k<<<NBLK(nSH64, T), T, 0, stream>>>(x1, w_sh1h, b_sh1h, h64, NSH, DEMB, DEMB);
  zero_k<<<NBLK(nSH64, T), T, 0, stream>>>(agg64, nSH64);
  scatter_add4_k<<<NBLK((long long)ESH * 16, T), T, 0, stream>>>(h64, eiSH, ESH, 16, agg64);
  mean_tanh_k<<<NBLK(nSH64, T), T, 0, stream>>>(agg64, cnt, x22, NSH, DEMB);

  lin_k<<<NBLK(nSH64, T), T, 0, stream>>>(x22, w_sh2h, b_sh2h, h64, NSH, DEMB, DEMB);
  zero_k<<<NBLK(nSH64, T), T, 0, stream>>>(agg64, nSH64);
  scatter_add4_k<<<NBLK((long long)ESH * 16, T), T, 0, stream>>>(h64, eiSH, ESH, 16, agg64);
  mean_tanh_k<<<NBLK(nSH64, T), T, 0, stream>>>(agg64, cnt, x66, NSH, DEMB);

  avg3_k<<<NBLK(nSH64, T), T, 0, stream>>>(x1, x22, x66, x99, nSH64);
  lin_k<<<NBLK(nSH256, T), T, 0, stream>>>(x99, w_mlp1h, b_mlp1h, y99, NSH, DEMB, HDIM);
  bn_stats_k<<<HDIM, 256, 0, stream>>>(y99, NSH, HDIM, bmean, brstd);
  bn_apply_k<<<NBLK(nSH256, T), T, 0, stream>>>(y99, bmean, brstd, g_bn1h, be_bn1h, nSH256, HDIM, 0);

  // ---- SS graph (sum aggregation) ----
  const long long nSS256 = (long long)NSS * HDIM;
  gather_rows_k<<<NBLK((long long)NSS * DEMB, T), T, 0, stream>>>(emb, xSS, xss0, NSS, DEMB);
  lin_k<<<NBLK(nSS256, T), T, 0, stream>>>(xss0, w_ss, b_ss, hss, NSS, DEMB, HDIM);
  zero_k<<<NBLK(nSS256, T), T, 0, stream>>>(aggss, nSS256);
  scatter_add4_k<<<NBLK((long long)ESS * 64, T), T, 0, stream>>>(hss, eiSS, ESS, 64, aggss);
  tanh_inplace_k<<<NBLK(nSS256, T), T, 0, stream>>>(aggss, nSS256);

  // ---- HH graph (sum aggregation, concat KG one-hot) ----
  const long long nHH256 = (long long)NHH * HDIM;
  concat_hh_k<<<NBLK((long long)NHH * 91, T), T, 0, stream>>>(emb, xHH, kg, xhh0, NHH);
  lin_k<<<NBLK(nHH256, T), T, 0, stream>>>(xhh0, w_hh, b_hh, hhh, NHH, 91, HDIM);
  zero_k<<<NBLK(nHH256, T), T, 0, stream>>>(agghh, nHH256);
  scatter_add4_k<<<NBLK((long long)EHH * 64, T), T, 0, stream>>>(hhh, eiHH, EHH, 64, agghh);
  tanh_inplace_k<<<NBLK(nHH256, T), T, 0, stream>>>(agghh, nHH256);

  // es = y9[:390] + x_ss1 ; eh = y99[390:] + x_hh1
  add_k<<<NBLK(nSS256, T), T, 0, stream>>>(y9, aggss, es, nSS256);
  add_k<<<NBLK(nHH256, T), T, 0, stream>>>(y99 + (long long)NSS * HDIM, agghh, eh, nHH256);

  // ---- big GEMMs (WMMA f32 16x16x4, 16x64 strip per wave) ----
  rowsum_k<<<NBLK(NB, 8), 256, 0, stream>>>(presc, psum, NB, NSS);

  {  // e0 = (prescription @ es) / psum[row]   [20000 x 256]
    long long waves = (long long)(NB / 16) * (HDIM / 64);
    gemm_scaled_k<<<NBLK(waves * 32, 256), 256, 0, stream>>>(presc, es, psum, e0, NB, HDIM, NSS);
  }
  {  // y2 = e0 @ w_mlp + b_mlp   [20000 x 256]
    long long waves = (long long)(NB / 16) * (HDIM / 64);
    gemm_bias_k<<<NBLK(waves * 32, 256), 256, 0, stream>>>(e0, w_mlp, b_mlp, y2, NB, HDIM, HDIM);
  }
  bn_stats_k<<<HDIM, 256, 0, stream>>>(y2, NB, HDIM, bmean, brstd);
  bn_apply_k<<<NBLK((long long)NB * HDIM, T), T, 0, stream>>>(y2, bmean, brstd, g_si, be_si,
                                                              (long long)NB * HDIM, HDIM, 1);
  {  // out = y2 @ eh^T   [20000 x 805]
    long long waves = (long long)(NB / 16) * (((long long)NHH + 63) / 64);
    gemm_bt_k<<<NBLK(waves * 32, 256), 256, 0, stream>>>(y2, eh, out, NB, NHH, HDIM);
  }
}